// PhasedLSTMCell_12343736008984
// MI455X (gfx1250) — compile-verified
//
#include <hip/hip_runtime.h>
#include <cmath>

// PhasedLSTM cell, fused GEMM + gate epilogue for gfx1250 (MI455X).
// gates = X @ Wxh^T + b + Hx @ Whh^T, done with V_WMMA_F32_16X16X4_F32
// (fp32 matrix pipe, matches reference precision). Each workgroup computes
// a 64(batch) x 32(hidden) tile for ALL FOUR gates so the epilogue is fully
// in-register and the 64MB gates intermediate never exists in memory.

typedef __attribute__((ext_vector_type(2))) float v2f;
typedef __attribute__((ext_vector_type(8))) float v8f;

#define B_    4096
#define IN_   512
#define H_    1024
#define KTOT  (IN_ + H_)   // 1536
#define BM    64           // batch rows per block
#define BN    32           // hidden columns per block (x4 gates internally)
#define KT    32           // K step per LDS stage

#define LDA   (KT + 4)     // pad to dodge bank conflicts
#define LDB   (BN + 1)

__global__ __launch_bounds__(256, 2)
void plstm_fused_kernel(const float* __restrict__ x,    // [B, IN]
                        const float* __restrict__ tme,  // [B, 1]
                        const float* __restrict__ hx,   // [B, H]
                        const float* __restrict__ cx,   // [B, H]
                        const float* __restrict__ Wxh,  // [4H, IN]
                        const float* __restrict__ bxh,  // [4H]
                        const float* __restrict__ Whh,  // [4H, H]
                        const float* __restrict__ tau,  // [H]
                        const float* __restrict__ sph,  // [H]
                        const float* __restrict__ alpha_p,
                        const float* __restrict__ rho_p,
                        float* __restrict__ out)        // [2, B, H] = hy, cy
{
    __shared__ float As[BM][LDA];        // batch-rows x K chunk
    __shared__ float Bs[4][KT][LDB];     // per-gate: K x hidden-cols

    const int tid  = threadIdx.x;
    const int lane = tid & 31;
    const int wave = tid >> 5;
    const int wm   = wave & 3;   // 0..3 : 16-row band within 64 rows
    const int wn   = wave >> 2;  // 0..1 : 16-col band within 32 cols

    const int rowBase = blockIdx.x * BM;
    const int colBase = blockIdx.y * BN;

    v8f acc[4] = {};  // i, f, g, o accumulators (16x16 f32 tile each)

    const int kSteps = KTOT / KT;  // 48; K=512 boundary is KT-aligned
    for (int ks = 0; ks < kSteps; ++ks) {
        const int  kb  = ks * KT;
        const bool inX = kb < IN_;

        // ---- prefetch next A chunk (emits global_prefetch_b8) ----
        if (ks + 1 < kSteps) {
            const int  kb2  = kb + KT;
            const bool inX2 = kb2 < IN_;
            const int  r    = tid >> 3;   // reuse A mapping, 1 line per thread
            const float* pf = inX2 ? (x  + (size_t)(rowBase + r) * IN_ + kb2)
                                   : (hx + (size_t)(rowBase + r) * H_  + (kb2 - IN_));
            __builtin_prefetch(pf, 0, 0);
        }

        // ---- stage A: 64 rows x 32 k  (512 float4, 2 per thread) ----
#pragma unroll
        for (int j = 0; j < 2; ++j) {
            const int idx = tid + 256 * j;
            const int r   = idx >> 3;      // 0..63
            const int kq  = idx & 7;       // float4 slot within KT
            const float* src = inX ? (x  + (size_t)(rowBase + r) * IN_ + kb)
                                   : (hx + (size_t)(rowBase + r) * H_  + (kb - IN_));
            const float4 v = *(const float4*)(src + kq * 4);
            As[r][kq * 4 + 0] = v.x;
            As[r][kq * 4 + 1] = v.y;
            As[r][kq * 4 + 2] = v.z;
            As[r][kq * 4 + 3] = v.w;
        }

        // ---- stage B: 4 gates x 32 cols x 32 k (1024 float4, 4/thread),
        //      transposed into K-major LDS so B fragments read along K ----
#pragma unroll
        for (int j = 0; j < 4; ++j) {
            const int idx = tid + 256 * j;
            const int gi  = idx >> 8;       // gate 0..3
            const int rem = idx & 255;
            const int n   = rem >> 3;       // hidden col 0..31
            const int kq  = rem & 7;
            const int wrow = gi * H_ + colBase + n;
            const float* src = inX ? (Wxh + (size_t)wrow * IN_ + kb)
                                   : (Whh + (size_t)wrow * H_  + (kb - IN_));
            const float4 v = *(const float4*)(src + kq * 4);
            Bs[gi][kq * 4 + 0][n] = v.x;
            Bs[gi][kq * 4 + 1][n] = v.y;
            Bs[gi][kq * 4 + 2][n] = v.z;
            Bs[gi][kq * 4 + 3][n] = v.w;
        }
        __syncthreads();

        // ---- 8 x K4 steps x 4 gates of V_WMMA_F32_16X16X4_F32 ----
        // A 16x4 f32 layout: lanes 0-15 -> K={k0,k0+1}, lanes 16-31 -> K={k0+2,k0+3}
        const int arow = wm * 16 + (lane & 15);
        const int bcol = wn * 16 + (lane & 15);
        const int khalf = 2 * (lane >> 4);
#pragma unroll
        for (int k0 = 0; k0 < KT; k0 += 4) {
            const int kk = k0 + khalf;
            v2f a;
            a.x = As[arow][kk];
            a.y = As[arow][kk + 1];
#pragma unroll
            for (int g = 0; g < 4; ++g) {
                v2f b;
                b.x = Bs[g][kk][bcol];
                b.y = Bs[g][kk + 1][bcol];
                acc[g] = __builtin_amdgcn_wmma_f32_16x16x4_f32(
                    /*neg_a=*/false, a, /*neg_b=*/false, b,
                    /*c_mod=*/(short)0, acc[g],
                    /*reuse_a=*/false, /*reuse_b=*/false);
            }
        }
        __syncthreads();
    }

    // ---- fused phased-LSTM epilogue, fully in-register ----
    // C/D layout: VGPR r, lane -> M = r + 8*(lane>=16), N = lane&15
    const float alpha = alpha_p[0];
    const float rho   = rho_p[0];
    const int   hcol  = colBase + wn * 16 + (lane & 15);
    const float tauh  = tau[hcol];
    const float sh    = sph[hcol];
    const float bi    = bxh[0 * H_ + hcol];
    const float bf    = bxh[1 * H_ + hcol];
    const float bg    = bxh[2 * H_ + hcol];
    const float bo    = bxh[3 * H_ + hcol];

#pragma unroll
    for (int r = 0; r < 8; ++r) {
        const int row = rowBase + wm * 16 + r + 8 * (lane >> 4);

        // time gate k (stop_gradient irrelevant in fwd)
        const float t   = tme[row];
        const float phi = fmodf(t - sh, tauh) / tauh;
        float kg;
        if (phi < rho * 0.5f)      kg = 2.0f * phi / rho;
        else if (phi < rho)        kg = 2.0f - 2.0f * phi / rho;
        else                       kg = alpha * phi;

        const float iv = acc[0][r] + bi;
        const float fv = acc[1][r] + bf;
        const float gv = acc[2][r] + bg;
        const float ov = acc[3][r] + bo;

        const float ft = 1.0f / (1.0f + __expf(-(fv + 1.0f - kg)));
        const float it = (1.0f / (1.0f + __expf(-iv))) * kg;
        const float gt = tanhf(gv) * kg;
        const float ot = (1.0f / (1.0f + __expf(-ov))) * kg;

        const float cxv = cx[(size_t)row * H_ + hcol];
        const float cyv = ft * cxv + it * gt;
        const float hyv = ot * tanhf(cyv);

        out[(size_t)row * H_ + hcol]                     = hyv;  // hy
        out[(size_t)B_ * H_ + (size_t)row * H_ + hcol]   = cyv;  // cy
    }
}

extern "C" void kernel_launch(void* const* d_in, const int* in_sizes, int n_in,
                              void* d_out, int out_size, void* d_ws, size_t ws_size,
                              hipStream_t stream) {
    const float* x     = (const float*)d_in[0];   // input   [B*IN]
    const float* tme   = (const float*)d_in[1];   // time    [B]
    const float* hx    = (const float*)d_in[2];   // hx      [B*H]
    const float* cx    = (const float*)d_in[3];   // cx      [B*H]
    const float* Wxh   = (const float*)d_in[4];   // Wxh_w   [4H*IN]
    const float* bxh   = (const float*)d_in[5];   // Wxh_b   [4H]
    const float* Whh   = (const float*)d_in[6];   // Whh_w   [4H*H]
    const float* tau   = (const float*)d_in[7];   // tau     [H]
    const float* sph   = (const float*)d_in[8];   // s       [H]
    const float* alpha = (const float*)d_in[9];   // alpha   [1]
    const float* rho   = (const float*)d_in[10];  // rho     [1]
    float* out = (float*)d_out;                   // [hy | cy], 2*B*H floats

    dim3 grid(B_ / BM, H_ / BN);  // 64 x 32 = 2048 workgroups
    plstm_fused_kernel<<<grid, 256, 0, stream>>>(
        x, tme, hx, cx, Wxh, bxh, Whh, tau, sph, alpha, rho, out);
}